// RoIBBox_25048249270464
// MI455X (gfx1250) — compile-verified
//
#include <hip/hip_runtime.h>
#include <hip/hip_bf16.h>
#include <stdint.h>

// ---------------------------------------------------------------------------
// RPN RoI proposal head for MI455X (gfx1250).
//   B=64 batches, A=8649 anchors, PRE=1000 (pre-NMS top-k), POST=256.
// MI455X-driven design (compile-only reasoning):
//   * ~11 MB in / 0.3 MB out -> ~0.5us of HBM traffic @23.3 TB/s. The cost
//     centers are top-k selection and the sequential greedy NMS.
//   * 320 KB LDS per WGP -> per-batch top-k as ONE in-LDS bitonic sort of
//     16384 packed (score,~idx) 64-bit keys (128 KB LDS), one WG per batch.
//   * TDM tensor_load_to_lds (TENSORcnt) DMAs the probs row (34.6 KB) and
//     the NMS box tile (16 KB) into LDS instead of scalarized vector loads.
//   * NMS = bitmask algorithm: 1M IoU tests -> 1000x32 suppression bitmask
//     (128 KB LDS) fully parallel, then ONE wave32 does the exact greedy
//     reduction with the keep mask in registers (one chunk/lane, __shfl
//     broadcast per step). 4 barriers total instead of ~1000.
// No matrix ops exist in this workload -> no v_wmma by design; the CDNA5
// signal is the async-tensor (TDM) path + big-LDS single-pass algorithms.
// ---------------------------------------------------------------------------

#define BATCH      64
#define NANCH      8649
#define SORT_N     16384          // next pow2 >= NANCH
#define PRE_TOPN   1000
#define POST_TOPN  256
#define NTHREADS   1024
#define IOU_THR    0.8f

// ---- TDM plumbing ---------------------------------------------------------
#if defined(__has_builtin)
#if __has_builtin(__builtin_amdgcn_tensor_load_to_lds)
#define HAVE_TDM 1
#endif
#endif

typedef unsigned int u32x4 __attribute__((ext_vector_type(4)));
typedef int          i32x8 __attribute__((ext_vector_type(8)));
typedef int          i32x4 __attribute__((ext_vector_type(4)));

#if defined(HAVE_TDM)
// DMA one contiguous row of n f32 elements from global -> LDS byte offset.
// D# group0: count=1 | lds_addr | 57-bit global addr | type=2 ("image")
// D# group1: data_size=4B, tensor_dim0=tile_dim0=n, dim1=tile_dim1=1.
__device__ __forceinline__ void tdm_load_f32_row(const float* gp,
                                                 unsigned lds_byte_off, int n) {
    unsigned long long ga = (unsigned long long)(uintptr_t)gp;
    u32x4 g0;
    g0[0] = 1u;                                                // count=1, user mode
    g0[1] = __builtin_amdgcn_groupstaticsize() + lds_byte_off; // LDS byte address
    g0[2] = (unsigned)ga;                                      // global_addr[31:0]
    g0[3] = ((unsigned)(ga >> 32) & 0x01FFFFFFu) | (2u << 30); // [56:32] | type=2
    i32x8 g1;
    g1[0] = (int)(2u << 16);                                   // data_size = 4 bytes
    g1[1] = (int)(((unsigned)n & 0xFFFFu) << 16);              // tensor_dim0[15:0]
    g1[2] = (int)((((unsigned)n >> 16) & 0xFFFFu) | (1u << 16)); // dim0 hi | dim1 lo
    g1[3] = (int)(((unsigned)n & 0xFFFFu) << 16);              // dim1 hi=0 | tile_dim0
    g1[4] = 1;                                                 // tile_dim1=1, tile_dim2=0
    g1[5] = n;                                                 // tensor_dim0_stride
    g1[6] = 0;
    g1[7] = 0;
    i32x4 z4 = {0, 0, 0, 0};
#if defined(__clang_major__) && (__clang_major__ >= 23)
    i32x8 z8 = {0, 0, 0, 0, 0, 0, 0, 0};
    __builtin_amdgcn_tensor_load_to_lds(g0, g1, z4, z4, z8, 0);
#else
    __builtin_amdgcn_tensor_load_to_lds(g0, g1, z4, z4, 0);
#endif
}
#endif

// ===========================================================================
// Kernel 1: per-batch top-1000 (full in-LDS bitonic sort) + bbox decode.
// LDS (dynamic, 165,888 B): keys u64[16384] @0 | probs f32[8704] @131072
// ===========================================================================
#define K1_KEY_BYTES   (SORT_N * 8)
#define K1_SMEM_BYTES  (K1_KEY_BYTES + 8704 * 4)

__global__ __launch_bounds__(NTHREADS)
void rpn_topk_decode_kernel(const float* __restrict__ probs,
                            const float* __restrict__ deltas,
                            const float* __restrict__ anchors,
                            float* __restrict__ top_score,   // [B,PRE]
                            float* __restrict__ top_boxes)   // [B,PRE,4]
{
    extern __shared__ unsigned char smem[];
    unsigned long long* keys = (unsigned long long*)smem;
    float* sprobs = (float*)(smem + K1_KEY_BYTES);

    const int b = blockIdx.x;
    const int t = threadIdx.x;
    const float* gp = probs + (size_t)b * NANCH;

#if defined(HAVE_TDM)
    if (t == 0) tdm_load_f32_row(gp, (unsigned)K1_KEY_BYTES, NANCH);
    // While the DMA is in flight, initialize the sort pad region.
    for (int i = t; i < SORT_N; i += NTHREADS)
        if (i >= NANCH) keys[i] = 0ull;                        // sorts to the bottom
    if (t == 0) __builtin_amdgcn_s_wait_tensorcnt(0);          // issuing wave waits
    __syncthreads();
#else
    for (int i = t; i < NANCH; i += NTHREADS) sprobs[i] = gp[i];
    for (int i = t; i < SORT_N; i += NTHREADS)
        if (i >= NANCH) keys[i] = 0ull;
    __syncthreads();
#endif

    // Pack keys: (score bits << 32) | ~idx -> descending sort reproduces
    // jax.lax.top_k ordering (score desc, index asc on ties); scores >= 0 so
    // their IEEE bits are order-preserving as u32.
    for (int i = t; i < NANCH; i += NTHREADS) {
        unsigned sb = __float_as_uint(sprobs[i]);
        keys[i] = ((unsigned long long)sb << 32) | (unsigned)(~i);
    }
    __syncthreads();

    // In-LDS bitonic sort, descending, N=16384 (105 stages).
    for (int k = 2; k <= SORT_N; k <<= 1) {
        for (int j = k >> 1; j > 0; j >>= 1) {
            for (int i = t; i < SORT_N; i += NTHREADS) {
                int l = i ^ j;
                if (l > i) {
                    bool desc = ((i & k) == 0);
                    unsigned long long a = keys[i];
                    unsigned long long c = keys[l];
                    if ((a < c) == desc) { keys[i] = c; keys[l] = a; }
                }
            }
            __syncthreads();
        }
    }

    // Emit top-1000 scores and decoded boxes (deltas * variances -> bbox).
    if (t < PRE_TOPN) {
        unsigned long long kv = keys[t];
        float sc = __uint_as_float((unsigned)(kv >> 32));
        int idx  = (int)(~(unsigned)kv);

        const float* an = anchors + (size_t)idx * 4;
        const float* dl = deltas + ((size_t)b * NANCH + (size_t)idx) * 4;
        float a0 = an[0], a1 = an[1], a2 = an[2], a3 = an[3];
        float ah = a2 - a0, aw = a3 - a1;
        float acy = a0 + 0.5f * ah, acx = a1 + 0.5f * aw;
        float d0 = dl[0] * 0.1f, d1 = dl[1] * 0.1f;
        float d2 = dl[2] * 0.2f, d3 = dl[3] * 0.2f;
        float h  = expf(d2) * ah;
        float w  = expf(d3) * aw;
        float cy = d0 * ah + acy;
        float cx = d1 * aw + acx;

        float* ob = top_boxes + ((size_t)b * PRE_TOPN + t) * 4;
        ob[0] = cy - 0.5f * h;
        ob[1] = cx - 0.5f * w;
        ob[2] = cy + 0.5f * h;
        ob[3] = cx + 0.5f * w;
        top_score[(size_t)b * PRE_TOPN + t] = sc;
    }
}

// ===========================================================================
// Kernel 2: bitmask NMS + compaction.
// LDS (dynamic, 152,132 B):
//   bx f32[1000*4]      @      0   (16,000 B)  <- TDM DMA target
//   area f32[1000]      @ 16,000   ( 4,000 B)
//   supmat u32[1000*32] @ 20,000   (128,000 B) suppression bitmask rows
//   keepw u32[32]       @148,000   (   128 B)  final keep mask
//   rank  i32[1000]     @148,128   ( 4,000 B)
//   total i32[1]        @152,128
// ===========================================================================
#define K2_BX_OFF    0
#define K2_AREA_OFF  16000
#define K2_SUP_OFF   20000
#define K2_KEEP_OFF  148000
#define K2_RANK_OFF  148128
#define K2_TOT_OFF   152128
#define K2_SMEM_BYTES (K2_TOT_OFF + 4)

__global__ __launch_bounds__(NTHREADS)
void rpn_nms_kernel(const float* __restrict__ top_boxes,   // [B,PRE,4]
                    const float* __restrict__ top_score,   // [B,PRE]
                    float* __restrict__ out_boxes,         // [B,POST,4]
                    float* __restrict__ out_scores)        // [B,POST]
{
    extern __shared__ unsigned char smem[];
    float*    bx     = (float*)(smem + K2_BX_OFF);
    float*    area   = (float*)(smem + K2_AREA_OFF);
    unsigned* supmat = (unsigned*)(smem + K2_SUP_OFF);
    unsigned* keepw  = (unsigned*)(smem + K2_KEEP_OFF);
    int*      rank   = (int*)(smem + K2_RANK_OFF);
    int*      total  = (int*)(smem + K2_TOT_OFF);

    const int b = blockIdx.x;
    const int t = threadIdx.x;
    const float* gb = top_boxes + (size_t)b * PRE_TOPN * 4;
    const float* ts = top_score + (size_t)b * PRE_TOPN;

    __builtin_prefetch(ts, 0, 1);  // global_prefetch_b8: warm GL2 for the scores

#if defined(HAVE_TDM)
    if (t == 0) {
        tdm_load_f32_row(gb, (unsigned)K2_BX_OFF, PRE_TOPN * 4);  // 16 KB DMA
        __builtin_amdgcn_s_wait_tensorcnt(0);
    }
    __syncthreads();
    for (int i = t; i < PRE_TOPN; i += NTHREADS)
        area[i] = (bx[i * 4 + 2] - bx[i * 4 + 0]) * (bx[i * 4 + 3] - bx[i * 4 + 1]);
#else
    for (int i = t; i < PRE_TOPN; i += NTHREADS) {
        float y1 = gb[i * 4 + 0], x1 = gb[i * 4 + 1];
        float y2 = gb[i * 4 + 2], x2 = gb[i * 4 + 3];
        bx[i * 4 + 0] = y1; bx[i * 4 + 1] = x1;
        bx[i * 4 + 2] = y2; bx[i * 4 + 3] = x2;
        area[i] = (y2 - y1) * (x2 - x1);
    }
#endif
    __syncthreads();

    // Phase 1 (parallel, barrier-free): suppression bitmask.
    //   supmat[i][c] bit jj set <=> j=32c+jj, j>i, iou(i,j) > THR
    for (int w = t; w < PRE_TOPN * 32; w += NTHREADS) {
        int i  = w >> 5;
        int j0 = (w & 31) << 5;
        float iy1 = bx[i * 4 + 0], ix1 = bx[i * 4 + 1];
        float iy2 = bx[i * 4 + 2], ix2 = bx[i * 4 + 3];
        float ia  = area[i];
        unsigned bits = 0u;
#pragma unroll
        for (int jj = 0; jj < 32; ++jj) {
            int j = j0 + jj;
            if (j < PRE_TOPN && j > i) {
                float yy1 = fmaxf(iy1, bx[j * 4 + 0]);
                float xx1 = fmaxf(ix1, bx[j * 4 + 1]);
                float yy2 = fminf(iy2, bx[j * 4 + 2]);
                float xx2 = fminf(ix2, bx[j * 4 + 3]);
                float inter = fmaxf(yy2 - yy1, 0.0f) * fmaxf(xx2 - xx1, 0.0f);
                float uni   = ia + area[j] - inter;
                if (inter / fmaxf(uni, 1e-8f) > IOU_THR) bits |= (1u << jj);
            }
        }
        supmat[w] = bits;
    }
    __syncthreads();

    // Phase 2 (one wave32): exact greedy reduction, keep mask in registers.
    //   lane c owns keep bits j in [32c, 32c+32); bit-i broadcast via shfl.
    if (t < 32) {
        unsigned keepc = 0xFFFFFFFFu;
        for (int i = 0; i < PRE_TOPN; ++i) {
            unsigned chunk = __shfl(keepc, i >> 5, 32);   // uniform broadcast
            if ((chunk >> (i & 31)) & 1u)                 // box i survived
                keepc &= ~supmat[i * 32 + t];             // rows only hold j>i bits
        }
        keepw[t] = keepc;
    }
    __syncthreads();

    // Serial rank compaction (1000 steps on one thread — negligible).
    if (t == 0) {
        int c = 0;
        for (int i = 0; i < PRE_TOPN; ++i) {
            rank[i] = c;
            c += (int)((keepw[i >> 5] >> (i & 31)) & 1u);
        }
        *total = c;
    }
    __syncthreads();

    // Scatter kept boxes (clipped) + scores; zero-fill the invalid tail.
    for (int i = t; i < PRE_TOPN; i += NTHREADS) {
        if (((keepw[i >> 5] >> (i & 31)) & 1u) && rank[i] < POST_TOPN) {
            int s = rank[i];
            float* ob = out_boxes + ((size_t)b * POST_TOPN + s) * 4;
            ob[0] = fminf(fmaxf(bx[i * 4 + 0], 0.0f), 1.0f);
            ob[1] = fminf(fmaxf(bx[i * 4 + 1], 0.0f), 1.0f);
            ob[2] = fminf(fmaxf(bx[i * 4 + 2], 0.0f), 1.0f);
            ob[3] = fminf(fmaxf(bx[i * 4 + 3], 0.0f), 1.0f);
            out_scores[(size_t)b * POST_TOPN + s] = ts[i];
        }
    }
    int tot = *total;
    if (tot > POST_TOPN) tot = POST_TOPN;
    for (int s = tot + t; s < POST_TOPN; s += NTHREADS) {
        float* ob = out_boxes + ((size_t)b * POST_TOPN + s) * 4;
        ob[0] = 0.0f; ob[1] = 0.0f; ob[2] = 0.0f; ob[3] = 0.0f;
        out_scores[(size_t)b * POST_TOPN + s] = 0.0f;
    }
}

extern "C" void kernel_launch(void* const* d_in, const int* in_sizes, int n_in,
                              void* d_out, int out_size, void* d_ws, size_t ws_size,
                              hipStream_t stream) {
    (void)in_sizes; (void)n_in; (void)out_size; (void)ws_size;
    const float* deltas  = (const float*)d_in[0];  // [B, A, 4]
    const float* probs   = (const float*)d_in[1];  // [B, A]
    // d_in[2] = gt_labels (unused by the reference output)
    const float* anchors = (const float*)d_in[3];  // [A, 4]

    float* out        = (float*)d_out;
    float* out_boxes  = out;                                   // [B,256,4]
    float* out_scores = out + (size_t)BATCH * POST_TOPN * 4;   // [B,256]

    float* top_boxes = (float*)d_ws;                                   // [B,PRE,4] 1.0 MB
    float* top_score = top_boxes + (size_t)BATCH * PRE_TOPN * 4;       // [B,PRE]  0.25 MB

    rpn_topk_decode_kernel<<<BATCH, NTHREADS, K1_SMEM_BYTES, stream>>>(
        probs, deltas, anchors, top_score, top_boxes);
    rpn_nms_kernel<<<BATCH, NTHREADS, K2_SMEM_BYTES, stream>>>(
        top_boxes, top_score, out_boxes, out_scores);
}